// StaticGraphConv_12309376270841
// MI455X (gfx1250) — compile-verified
//
#include <hip/hip_runtime.h>

// ---------------------------------------------------------------------------
// StaticGraphConv on MI455X (gfx1250, wave32).
//
//   K0: zero out[] and BN accumulators
//   K1: centroid grouped-linear + relu -> xc                    [tiny]
//   K2: TDM weight stage + per-edge message build (b64 gathers)
//       + grouped GEMM via V_WMMA_F32_16X16X4_F32 + BN partials [dominant]
//   K3: BN finalize                                             [tiny]
//   K4: normalize + relu + atomic segment-max (b128 loads)      [bandwidth]
//
// ws layout (floats): h[E*64] | xc[CENTERS*64] | bnacc[128] | scaleshift[128]
// ---------------------------------------------------------------------------

typedef float v2f __attribute__((ext_vector_type(2)));
typedef float v8f __attribute__((ext_vector_type(8)));
typedef unsigned int u32x4_t __attribute__((ext_vector_type(4)));
typedef int i32x4_t __attribute__((ext_vector_type(4)));
typedef int i32x8_t __attribute__((ext_vector_type(8)));

#define N_NODES  50000
#define N_EDGES  800000
#define DIM      64
#define OUT_C    64
#define CENTERS  1000
#define GROUPS   4
#define CIN      192           // 2*DIM + DC
#define KPG      48            // CIN / GROUPS
#define MPAD     12            // row stride 204: 8B-aligned, 16 rows -> 16 banks
#define BN_EPS   1e-5f

#define TPB      128           // 4 wave32
#define EPB      128           // edge stride per block (waves cover 4*16 = 64)

// amdgpu-toolchain (therock headers) => 6-arg tensor_load_to_lds builtin;
// ROCm 7.2 => 5-arg form.
#if defined(__has_include)
#  if __has_include(<hip/amd_detail/amd_gfx1250_TDM.h>)
#    define TDM_ARITY6 1
#  else
#    define TDM_ARITY6 0
#  endif
#else
#  define TDM_ARITY6 0
#endif

#if defined(__has_builtin)
#  if __has_builtin(__builtin_amdgcn_tensor_load_to_lds)
#    define HAVE_TDM 1
#  else
#    define HAVE_TDM 0
#  endif
#else
#  define HAVE_TDM 0
#endif

#if HAVE_TDM
// 1D contiguous copy of nelem32 dwords: global -> LDS via the Tensor Data
// Mover.  D# group0: flags | lds_addr | global_addr | type=2.  Group1:
// data_size=4B, tensor_dim0 = tile_dim0 = nelem32 (single row).
__device__ __forceinline__ void tdm_copy_dwords_to_lds(unsigned lds_off,
                                                       const void* gptr,
                                                       unsigned nelem32) {
    unsigned long long ga = (unsigned long long)(size_t)gptr;
    u32x4_t g0;
    g0[0] = 1u;                                              // count=1
    g0[1] = lds_off;                                         // lds_addr
    g0[2] = (unsigned)(ga & 0xffffffffu);                    // global_addr lo
    g0[3] = (unsigned)((ga >> 32) & 0x01ffffffu) | (2u << 30); // hi | type=2
    i32x8_t g1;
    g1[0] = (int)(2u << 16);                                 // data_size=4B
    g1[1] = (int)((nelem32 & 0xffffu) << 16);                // tensor_dim0 lo
    g1[2] = (int)(((nelem32 >> 16) & 0xffffu) | (1u << 16)); // dim0 hi|dim1=1
    g1[3] = (int)((nelem32 & 0xffffu) << 16);                // tile_dim0
    g1[4] = 1;                                               // tile_dim1=1
    g1[5] = (int)nelem32;                                    // dim0_stride lo
    g1[6] = 0;
    g1[7] = 0;
    i32x4_t z4 = {0, 0, 0, 0};
#if TDM_ARITY6
    i32x8_t z8 = {0, 0, 0, 0, 0, 0, 0, 0};
    __builtin_amdgcn_tensor_load_to_lds(g0, g1, z4, z4, z8, 0);
#else
    __builtin_amdgcn_tensor_load_to_lds(g0, g1, z4, z4, 0);
#endif
}
#endif

// --------------------------------------------------------------------------
__global__ void init_kernel(float* __restrict__ out, float* __restrict__ bnacc) {
    long i = (long)blockIdx.x * blockDim.x + threadIdx.x;
    if (i < (long)N_NODES * OUT_C) out[i] = 0.0f;
    if (i < 2 * OUT_C) bnacc[i] = 0.0f;
}

// --------------------------------------------------------------------------
__global__ void centroid_kernel(const float* __restrict__ xce,
                                const float* __restrict__ Wc,
                                const float* __restrict__ bc,
                                float* __restrict__ xc) {
    int i = blockIdx.x * blockDim.x + threadIdx.x;
    if (i >= CENTERS * OUT_C) return;
    int c = i >> 6, o = i & 63;
    int g = o >> 4, oo = o & 15;
    const float* w  = Wc + (g * 16) * 16 + oo;    // W_c[g][k][oo], k-stride 16
    const float* xr = xce + c * DIM + g * 16;
    float acc = bc[o];
#pragma unroll
    for (int k = 0; k < 16; ++k) acc = fmaf(xr[k], w[k * 16], acc);
    xc[i] = fmaxf(acc, 0.0f);
}

// --------------------------------------------------------------------------
__global__ void __launch_bounds__(TPB)
edge_kernel(const float* __restrict__ x,   const int* __restrict__ batch,
            const int* __restrict__ src,   const int* __restrict__ dst,
            const float* __restrict__ xc,  const float* __restrict__ Wv,
            const float* __restrict__ bv,
            float* __restrict__ h,         float* __restrict__ bnacc) {
    __shared__ float sM[4][16][CIN + MPAD];       // 51 KB message staging
    __shared__ float sWv[GROUPS * KPG * 16];      // 12 KB: W_v[g][k][n]
    __shared__ float sBias[OUT_C];
    __shared__ float sSum[OUT_C];
    __shared__ float sSq[OUT_C];

    const int tid  = threadIdx.x;
    const int wave = tid >> 5;
    const int lane = tid & 31;
    const int e0   = blockIdx.x * EPB + wave * 16;

#if HAVE_TDM
    if (wave == 0) {
        tdm_copy_dwords_to_lds((unsigned)(size_t)&sWv[0], Wv, GROUPS * KPG * 16);
        __builtin_amdgcn_s_wait_tensorcnt(0);
    }
#else
    for (int i = tid; i < GROUPS * KPG * 16; i += TPB) sWv[i] = Wv[i];
#endif
    if (tid < OUT_C) { sBias[tid] = bv[tid]; sSum[tid] = 0.f; sSq[tid] = 0.f; }

    // Prefetch this wave's gathered node rows (global_prefetch_b8).
    {
        const int pe = e0 + (lane & 15);
        const int pn = (lane < 16) ? dst[pe] : src[pe];
        __builtin_prefetch(&x[(long)pn * DIM], 0, 0);
    }

    // Stage messages: lane covers features [2*lane, 2*lane+1] -> b64 traffic.
    const int f2 = lane << 1;
    for (int e = 0; e < 16; ++e) {
        const int eg = e0 + e;
        const int s = src[eg], d = dst[eg];
        const float2 xi = *(const float2*)&x[(long)d * DIM + f2];
        const float2 xj = *(const float2*)&x[(long)s * DIM + f2];
        const int   c   = batch[s];
        const float2 yj = *(const float2*)&xc[c * OUT_C + f2];
        float* row = &sM[wave][e][0];
        *(float2*)&row[f2]       = xi;
        *(float2*)&row[64 + f2]  = make_float2(xj.x - xi.x, xj.y - xi.y);
        *(float2*)&row[128 + f2] = yj;
    }
    __syncthreads();

    // f32 16x16x4 fragment layouts (ISA 7.12.2):
    //   A 16x4: lanes 0-15 M=lane, v0=K0 v1=K1; lanes 16-31 M=lane-16, K2/K3
    //   B  4x16: mirrored K striping, N = lane&15
    //   C/D:     VGPR r -> M = r + 8*(lane>=16), N = lane&15
    const int mrow  = lane & 15;
    const int khalf = (lane >> 4) << 1;   // 0 or 2
    const int n     = lane & 15;
    const int rbase = (lane >> 4) << 3;   // 0 or 8

    for (int g = 0; g < GROUPS; ++g) {
        v8f acc;
        const float bias = sBias[g * 16 + n];
#pragma unroll
        for (int r = 0; r < 8; ++r) acc[r] = bias;

#pragma unroll
        for (int k = 0; k < KPG; k += 4) {
            v2f a, b;
            a.x = sM[wave][mrow][g * KPG + k + khalf + 0];
            a.y = sM[wave][mrow][g * KPG + k + khalf + 1];
            b.x = sWv[(g * KPG + k + khalf + 0) * 16 + n];
            b.y = sWv[(g * KPG + k + khalf + 1) * 16 + n];
            // (neg_a, A, neg_b, B, c_mod, C, reuse_a, reuse_b)
            acc = __builtin_amdgcn_wmma_f32_16x16x4_f32(
                      false, a, false, b, (short)0, acc, false, false);
        }

        float psum = 0.f, psq = 0.f;
#pragma unroll
        for (int r = 0; r < 8; ++r) {
            const float v = acc[r];
            h[(long)(e0 + rbase + r) * OUT_C + g * 16 + n] = v;
            psum += v;
            psq  = fmaf(v, v, psq);
        }
        atomicAdd(&sSum[g * 16 + n], psum);
        atomicAdd(&sSq[g * 16 + n], psq);
    }
    __syncthreads();

    if (tid < OUT_C) {
        atomicAdd(&bnacc[tid], sSum[tid]);
        atomicAdd(&bnacc[OUT_C + tid], sSq[tid]);
    }
}

// --------------------------------------------------------------------------
__global__ void bn_finalize(const float* __restrict__ bnacc,
                            const float* __restrict__ gamma,
                            const float* __restrict__ beta,
                            float* __restrict__ ss) {
    int t = threadIdx.x;
    if (t < OUT_C) {
        const float inv = 1.0f / (float)N_EDGES;
        const float mu  = bnacc[t] * inv;
        const float var = bnacc[OUT_C + t] * inv - mu * mu;
        const float sc  = gamma[t] * rsqrtf(var + BN_EPS);
        ss[t]          = sc;
        ss[OUT_C + t]  = beta[t] - mu * sc;    // shift
    }
}

// --------------------------------------------------------------------------
// One thread per 4 channels: b128 load of h, then 4 atomic maxes.  v >= 0
// after relu so uint bit-pattern compare == float compare; out pre-zeroed.
__global__ void scatter_max(const float* __restrict__ h,
                            const int* __restrict__ dst,
                            const float* __restrict__ ss,
                            float* __restrict__ out) {
    __shared__ float sScale[OUT_C], sShift[OUT_C];
    if (threadIdx.x < OUT_C) {
        sScale[threadIdx.x] = ss[threadIdx.x];
        sShift[threadIdx.x] = ss[OUT_C + threadIdx.x];
    }
    __syncthreads();
    long q = (long)blockIdx.x * blockDim.x + threadIdx.x;   // quad index
    if (q >= (long)N_EDGES * (OUT_C / 4)) return;
    const long e   = q >> 4;                 // 16 quads per edge
    const int  ch0 = (int)((q & 15) << 2);
    const float4 hv = *(const float4*)&h[q << 2];
    unsigned int* ob = (unsigned int*)&out[(long)dst[e] * OUT_C + ch0];
    const float v0 = fmaxf(fmaf(hv.x, sScale[ch0 + 0], sShift[ch0 + 0]), 0.0f);
    const float v1 = fmaxf(fmaf(hv.y, sScale[ch0 + 1], sShift[ch0 + 1]), 0.0f);
    const float v2 = fmaxf(fmaf(hv.z, sScale[ch0 + 2], sShift[ch0 + 2]), 0.0f);
    const float v3 = fmaxf(fmaf(hv.w, sScale[ch0 + 3], sShift[ch0 + 3]), 0.0f);
    atomicMax(ob + 0, __float_as_uint(v0));
    atomicMax(ob + 1, __float_as_uint(v1));
    atomicMax(ob + 2, __float_as_uint(v2));
    atomicMax(ob + 3, __float_as_uint(v3));
}

// --------------------------------------------------------------------------
extern "C" void kernel_launch(void* const* d_in, const int* in_sizes, int n_in,
                              void* d_out, int out_size, void* d_ws, size_t ws_size,
                              hipStream_t stream) {
    const float* x        = (const float*)d_in[0];
    const int*   batch    = (const int*)  d_in[1];
    const int*   eidx     = (const int*)  d_in[2];   // [2, E]: src then dst
    const float* x_center = (const float*)d_in[3];
    // d_in[4] = batch_center (unused by reference)
    const float* Wc    = (const float*)d_in[5];
    const float* bc    = (const float*)d_in[6];
    const float* Wv    = (const float*)d_in[7];
    const float* bv    = (const float*)d_in[8];
    const float* gamma = (const float*)d_in[9];
    const float* beta  = (const float*)d_in[10];
    float* out = (float*)d_out;

    float* ws    = (float*)d_ws;
    float* h     = ws;                              // E*64 floats (~205 MB)
    float* xc    = h + (size_t)N_EDGES * OUT_C;     // CENTERS*64
    float* bnacc = xc + CENTERS * OUT_C;            // 128
    float* ss    = bnacc + 2 * OUT_C;               // 128

    const int* srce = eidx;
    const int* dste = eidx + N_EDGES;

    const long outTot = (long)N_NODES * OUT_C;
    init_kernel<<<(int)((outTot + 255) / 256), 256, 0, stream>>>(out, bnacc);

    centroid_kernel<<<(CENTERS * OUT_C + 255) / 256, 256, 0, stream>>>(
        x_center, Wc, bc, xc);

    edge_kernel<<<N_EDGES / EPB, TPB, 0, stream>>>(
        x, batch, srce, dste, xc, Wv, bv, h, bnacc);

    bn_finalize<<<1, 64, 0, stream>>>(bnacc, gamma, beta, ss);

    const long qTot = (long)N_EDGES * (OUT_C / 4);
    scatter_max<<<(int)((qTot + 255) / 256), 256, 0, stream>>>(h, dste, ss, out);
}